// MambaFFNet_47390669144439
// MI455X (gfx1250) — compile-verified
//
#include <hip/hip_runtime.h>
#include <hip/hip_bf16.h>

typedef __attribute__((ext_vector_type(2))) float v2f;
typedef __attribute__((ext_vector_type(8))) float v8f;

#define NCLS   10
#define BATCH  4
#define LSEQ   256
#define DIN    138
#define DPAD   144      // DIN padded to multiple of 16 (and of 4 for K-steps)
#define DMODEL 256
#define DINNER 512
#define DSTATE 16
#define DTRANK 16
#define ROWS   (BATCH*LSEQ)   // 1024 rows per class

// ---------------------------------------------------------------------------
// K0: max|x| over the whole input, plus zero the goodness accumulator.
// ---------------------------------------------------------------------------
__global__ void k_maxabs(const float* __restrict__ x, float* __restrict__ val,
                         float* __restrict__ g) {
    __shared__ float sm[256];
    int tid = threadIdx.x;
    float m = 0.f;
    for (int i = tid; i < BATCH * LSEQ * DIN; i += 256) m = fmaxf(m, fabsf(x[i]));
    sm[tid] = m; __syncthreads();
    for (int s = 128; s > 0; s >>= 1) {
        if (tid < s) sm[tid] = fmaxf(sm[tid], sm[tid + s]);
        __syncthreads();
    }
    if (tid == 0) val[0] = sm[0];
    if (tid < NCLS * BATCH) g[tid] = 0.f;
}

// ---------------------------------------------------------------------------
// Build x_c padded to DPAD: first NCLS cols = one-hot(class)*val, then x, then 0.
// Layout: xpad[class][row][DPAD], row = b*LSEQ + t
// ---------------------------------------------------------------------------
__global__ void k_build_x(const float* __restrict__ x, const float* __restrict__ val,
                          float* __restrict__ xpad) {
    size_t idx = (size_t)blockIdx.x * blockDim.x + threadIdx.x;
    size_t total = (size_t)NCLS * ROWS * DPAD;
    if (idx >= total) return;
    int col = (int)(idx % DPAD);
    size_t rc = idx / DPAD;
    int row = (int)(rc % ROWS);
    int cls = (int)(rc / ROWS);
    float v;
    if (col < NCLS)      v = (col == cls) ? val[0] : 0.f;
    else if (col < DIN)  v = x[(size_t)row * DIN + col];
    else                 v = 0.f;
    xpad[idx] = v;
}

// Zero-pad proj_w (138x256) -> (144x256)
__global__ void k_pad_w(const float* __restrict__ w, float* __restrict__ wpad) {
    int idx = blockIdx.x * blockDim.x + threadIdx.x;
    if (idx >= DPAD * DMODEL) return;
    int r = idx / DMODEL;
    wpad[idx] = (r < DIN) ? w[idx] : 0.f;
}

// ---------------------------------------------------------------------------
// Fully-static fp32 WMMA GEMM: D[cls] = A[cls] @ Bw (+bias) (+addsrc[cls]) (act)
// One wave per 16x16 tile, K-loop in steps of 4 with V_WMMA_F32_16X16X4_F32.
// All shapes/strides are template constants so A loads are a single b64 with
// immediate offsets and B loads fold k*LDB into instruction offsets.
//
// A layout (16x4, v2f): lanes 0-15 hold K=k0,k0+1 of row M=lane; lanes 16-31
// hold K=k0+2,k0+3 of row M=lane-16. B layout (4x16, v2f): lanes 0-15 hold
// rows k0,k0+1 at N=lane; lanes 16-31 hold rows k0+2,k0+3 at N=lane-16.
// C/D (16x16, v8f): vgpr r = rows M=r (lanes 0-15) and M=r+8 (lanes 16-31).
// ACT: 0 = none, 1 = softplus.
// ---------------------------------------------------------------------------
template <int K, int LDA, int LDB, int LDD, int ACT, bool HASBIAS, bool HASADD>
__global__ void k_gemm(const float* __restrict__ A,
                       const float* __restrict__ Bw,
                       const float* __restrict__ bias,
                       const float* __restrict__ addsrc,
                       float* __restrict__ D) {
    int lane = threadIdx.x;
    int lrow = lane & 15;
    int hi   = lane >> 4;            // 0 or 1
    int n0   = blockIdx.x * 16;
    int m0   = blockIdx.y * 16;
    int cls  = blockIdx.z;

    const float* Ap = A + (size_t)cls * (ROWS * LDA) + (size_t)(m0 + lrow) * LDA + hi * 2;
    const float* Bp = Bw + (size_t)(hi * 2) * LDB + n0 + lrow;

    v8f acc = {};
#pragma unroll 8
    for (int k = 0; k < K; k += 4) {
        v2f a = *(const v2f*)(Ap + k);          // 8B-aligned for all instantiations
        v2f b;
        b.x = Bp[k * LDB];
        b.y = Bp[(k + 1) * LDB];
        acc = __builtin_amdgcn_wmma_f32_16x16x4_f32(false, a, false, b,
                                                    (short)0, acc, false, false);
    }

    int col = n0 + lrow;
    float bv = HASBIAS ? bias[col] : 0.f;
#pragma unroll
    for (int r = 0; r < 8; r++) {
        int row = m0 + hi * 8 + r;
        float v = acc[r] + bv;
        if (HASADD) v += addsrc[(size_t)cls * (ROWS * LDD) + (size_t)row * LDD + col];
        if (ACT == 1) v = (v > 20.f) ? v : log1pf(expf(v));   // softplus
        D[(size_t)cls * (ROWS * LDD) + (size_t)row * LDD + col] = v;
    }
}

// ---------------------------------------------------------------------------
// LayerNorm over last dim (DMODEL), one block (256 threads) per row.
// ---------------------------------------------------------------------------
__global__ void k_layernorm(const float* __restrict__ h, const float* __restrict__ w,
                            const float* __restrict__ b, float* __restrict__ hn) {
    __shared__ float s1[256], s2[256];
    int tid = threadIdx.x;
    size_t row = blockIdx.x;
    float v = h[row * DMODEL + tid];
    s1[tid] = v; s2[tid] = v * v; __syncthreads();
    for (int s = 128; s > 0; s >>= 1) {
        if (tid < s) { s1[tid] += s1[tid + s]; s2[tid] += s2[tid + s]; }
        __syncthreads();
    }
    float m   = s1[0] / DMODEL;
    float var = s2[0] / DMODEL - m * m;
    float r   = rsqrtf(var + 1e-5f);
    hn[row * DMODEL + tid] = (v - m) * r * w[tid] + b[tid];
}

// ---------------------------------------------------------------------------
// Causal depthwise conv (D_CONV=4) + SiLU. x_in = xz[..., :DINNER].
// xz row stride = 2*DINNER.
// ---------------------------------------------------------------------------
__global__ void k_conv_silu(const float* __restrict__ xz, const float* __restrict__ cw,
                            const float* __restrict__ cb, float* __restrict__ xc) {
    size_t idx = (size_t)blockIdx.x * blockDim.x + threadIdx.x;
    size_t total = (size_t)NCLS * ROWS * DINNER;
    if (idx >= total) return;
    int d = (int)(idx % DINNER);
    size_t r = idx / DINNER;
    int t = (int)(r % LSEQ);
    size_t cbi = r / LSEQ;                           // class*BATCH + b
    const float* xin = xz + cbi * (size_t)LSEQ * 2 * DINNER + d;
    float acc = cb[d];
#pragma unroll
    for (int j = 0; j < 4; j++) {
        int tt = t - 3 + j;
        if (tt >= 0) acc += cw[d * 4 + j] * xin[(size_t)tt * 2 * DINNER];
    }
    xc[idx] = acc / (1.f + expf(-acc));              // SiLU
}

// ---------------------------------------------------------------------------
// Selective scan: one thread per (class, b, d_inner); 16-state register scan.
// Fuses +xc*D_p and *silu(z).
// ---------------------------------------------------------------------------
__global__ void k_scan(const float* __restrict__ delta, const float* __restrict__ xc,
                       const float* __restrict__ xdbl, const float* __restrict__ xz,
                       const float* __restrict__ Alog, const float* __restrict__ Dp,
                       float* __restrict__ y) {
    int idx = blockIdx.x * blockDim.x + threadIdx.x;
    if (idx >= NCLS * BATCH * DINNER) return;
    int d  = idx % DINNER;
    int cb = idx / DINNER;                           // class*BATCH + b

    float Arow[DSTATE], hstate[DSTATE];
#pragma unroll
    for (int n = 0; n < DSTATE; n++) { Arow[n] = -expf(Alog[d * DSTATE + n]); hstate[n] = 0.f; }
    float dpv = Dp[d];
    size_t rowbase = (size_t)cb * LSEQ;

    for (int t = 0; t < LSEQ; t++) {
        size_t row = rowbase + t;
        float dt  = delta[row * DINNER + d];
        float xv  = xc[row * DINNER + d];
        const float* bc = xdbl + row * (DTRANK + 2 * DSTATE);   // [dr(16)|B(16)|C(16)]
        float dtx = dt * xv;
        float acc = 0.f;
#pragma unroll
        for (int n = 0; n < DSTATE; n++) {
            float dA = expf(dt * Arow[n]);
            hstate[n] = dA * hstate[n] + dtx * bc[DTRANK + n];
            acc += hstate[n] * bc[DTRANK + DSTATE + n];
        }
        acc += xv * dpv;
        float z = xz[row * 2 * DINNER + DINNER + d];
        acc *= z / (1.f + expf(-z));                 // * silu(z)
        y[row * DINNER + d] = acc;
    }
}

// One block per (class,b): g[cls*BATCH+b] += mean(out^2)
__global__ void k_goodness(const float* __restrict__ out, float* __restrict__ g) {
    __shared__ float sm[256];
    int tid = threadIdx.x;
    int cb = blockIdx.x;
    const float* p = out + (size_t)cb * LSEQ * DMODEL;
    float s = 0.f;
    for (int i = tid; i < LSEQ * DMODEL; i += 256) { float v = p[i]; s += v * v; }
    sm[tid] = s; __syncthreads();
    for (int st = 128; st > 0; st >>= 1) {
        if (tid < st) sm[tid] += sm[tid + st];
        __syncthreads();
    }
    if (tid == 0) g[cb] += sm[0] / (float)(LSEQ * DMODEL);
}

// h = out / (||out||_row + 1e-6), one block per row
__global__ void k_rownorm(const float* __restrict__ out, float* __restrict__ h) {
    __shared__ float sm[256];
    int tid = threadIdx.x;
    size_t row = blockIdx.x;
    float v = out[row * DMODEL + tid];
    sm[tid] = v * v; __syncthreads();
    for (int s = 128; s > 0; s >>= 1) {
        if (tid < s) sm[tid] += sm[tid + s];
        __syncthreads();
    }
    float nrm = sqrtf(sm[0]) + 1e-6f;
    h[row * DMODEL + tid] = v / nrm;
}

__global__ void k_argmax(const float* __restrict__ g, int* __restrict__ outp) {
    int b = threadIdx.x;
    if (b >= BATCH) return;
    int best = 0; float bv = g[b];
    for (int c = 1; c < NCLS; c++) {
        float v = g[c * BATCH + b];
        if (v > bv) { bv = v; best = c; }
    }
    outp[b] = best;
}

// ---------------------------------------------------------------------------
extern "C" void kernel_launch(void* const* d_in, const int* in_sizes, int n_in,
                              void* d_out, int out_size, void* d_ws, size_t ws_size,
                              hipStream_t stream) {
    (void)in_sizes; (void)n_in; (void)out_size; (void)ws_size;
    const float* x       = (const float*)d_in[0];
    const float* proj_w  = (const float*)d_in[1];
    const float* proj_b  = (const float*)d_in[2];
    const float* norm_w  = (const float*)d_in[3];
    const float* norm_b  = (const float*)d_in[4];
    const float* in_w    = (const float*)d_in[5];
    const float* conv_w  = (const float*)d_in[6];
    const float* conv_b  = (const float*)d_in[7];
    const float* xproj_w = (const float*)d_in[8];
    const float* dt_w    = (const float*)d_in[9];
    const float* dt_b    = (const float*)d_in[10];
    const float* A_log   = (const float*)d_in[11];
    const float* D_p     = (const float*)d_in[12];
    const float* out_w   = (const float*)d_in[13];

    float* ws = (float*)d_ws;
    size_t off = 0;
    float* val  = ws + off; off += 64;
    float* g    = ws + off; off += 64;
    float* wpad = ws + off; off += (size_t)DPAD * DMODEL;
    float* xpad = ws + off; off += (size_t)NCLS * ROWS * DPAD;
    float* hbuf = ws + off; off += (size_t)NCLS * ROWS * DMODEL;
    float* hn   = ws + off; off += (size_t)NCLS * ROWS * DMODEL;
    float* xz   = ws + off; off += (size_t)NCLS * ROWS * 2 * DINNER;
    float* xc   = ws + off; off += (size_t)NCLS * ROWS * DINNER;
    float* xdbl = ws + off; off += (size_t)NCLS * ROWS * (DTRANK + 2 * DSTATE);
    float* dlt  = ws + off; off += (size_t)NCLS * ROWS * DINNER;
    float* ybuf = ws + off; off += (size_t)NCLS * ROWS * DINNER;
    float* outb = hn;   // hn is dead after the in-proj GEMM of each layer

    dim3 wave(32, 1, 1);

    k_maxabs<<<1, 256, 0, stream>>>(x, val, g);
    k_pad_w<<<(DPAD * DMODEL + 255) / 256, 256, 0, stream>>>(proj_w, wpad);
    {
        size_t tot = (size_t)NCLS * ROWS * DPAD;
        k_build_x<<<(unsigned)((tot + 255) / 256), 256, 0, stream>>>(x, val, xpad);
    }
    // h = x_c @ proj_w + proj_b                       (M=1024, N=256, K=144)
    k_gemm<DPAD, DPAD, DMODEL, DMODEL, 0, true, false>
        <<<dim3(DMODEL / 16, ROWS / 16, NCLS), wave, 0, stream>>>(
            xpad, wpad, proj_b, nullptr, hbuf);

    for (int l = 0; l < 2; l++) {
        k_layernorm<<<NCLS * ROWS, 256, 0, stream>>>(hbuf, norm_w + l * DMODEL,
                                                     norm_b + l * DMODEL, hn);
        // xz = hn @ in_w[l]                           (M=1024, N=1024, K=256)
        k_gemm<DMODEL, DMODEL, 2 * DINNER, 2 * DINNER, 0, false, false>
            <<<dim3(2 * DINNER / 16, ROWS / 16, NCLS), wave, 0, stream>>>(
                hn, in_w + (size_t)l * DMODEL * 2 * DINNER, nullptr, nullptr, xz);
        {
            size_t tot = (size_t)NCLS * ROWS * DINNER;
            k_conv_silu<<<(unsigned)((tot + 255) / 256), 256, 0, stream>>>(
                xz, conv_w + (size_t)l * DINNER * 4, conv_b + l * DINNER, xc);
        }
        // x_dbl = xc @ xproj_w[l]                     (M=1024, N=48, K=512)
        k_gemm<DINNER, DINNER, DTRANK + 2 * DSTATE, DTRANK + 2 * DSTATE, 0, false, false>
            <<<dim3((DTRANK + 2 * DSTATE) / 16, ROWS / 16, NCLS), wave, 0, stream>>>(
                xc, xproj_w + (size_t)l * DINNER * (DTRANK + 2 * DSTATE),
                nullptr, nullptr, xdbl);
        // delta = softplus(dr @ dt_w[l] + dt_b[l])    (M=1024, N=512, K=16, lda=48)
        k_gemm<DTRANK, DTRANK + 2 * DSTATE, DINNER, DINNER, 1, true, false>
            <<<dim3(DINNER / 16, ROWS / 16, NCLS), wave, 0, stream>>>(
                xdbl, dt_w + (size_t)l * DTRANK * DINNER, dt_b + l * DINNER,
                nullptr, dlt);
        k_scan<<<(NCLS * BATCH * DINNER + 255) / 256, 256, 0, stream>>>(
            dlt, xc, xdbl, xz, A_log + (size_t)l * DINNER * DSTATE, D_p + l * DINNER, ybuf);
        // out = y @ out_w[l] + h  (residual fused)    (M=1024, N=256, K=512)
        k_gemm<DINNER, DINNER, DMODEL, DMODEL, 0, false, true>
            <<<dim3(DMODEL / 16, ROWS / 16, NCLS), wave, 0, stream>>>(
                ybuf, out_w + (size_t)l * DINNER * DMODEL, nullptr, hbuf, outb);
        k_goodness<<<NCLS * BATCH, 256, 0, stream>>>(outb, g);
        k_rownorm<<<NCLS * ROWS, 256, 0, stream>>>(outb, hbuf);
    }
    k_argmax<<<1, 32, 0, stream>>>(g, (int*)d_out);
}